// BasicModel_25409026523724
// MI455X (gfx1250) — compile-verified
//
#include <hip/hip_runtime.h>
#include <math.h>

typedef float v2f __attribute__((ext_vector_type(2)));
typedef float v8f __attribute__((ext_vector_type(8)));

#define N_NODES 200000
#define NGRAPH  2000
#define NPG     100
#define NEDGE   2000000
#define NREL    3

// ---------------------------------------------------------------------------
// Dense transform: out = h (N x DIN, stride hs)  @  Wcat (DIN x 128)
// Wcat = [W0 | W1 | W2 | root], W_r = sum_b comp[r][b]*basis[b]
// Output stored as 4 planes hw[plane][n][32] (plane 3 = root transform).
// Uses V_WMMA_F32_16X16X4_F32 (f32 in/out -> exact precision match).
// A fragments (16 rows x DIN) are loaded ONCE into registers and reused
// across all 8 column tiles; B comes from LDS each step.
// ---------------------------------------------------------------------------
template <int DIN>
__global__ __launch_bounds__(256) void rgcn_transform(
    const float* __restrict__ h, int hstride,
    const float* __restrict__ basis, const float* __restrict__ comp,
    const float* __restrict__ root, float* __restrict__ hw)
{
    __shared__ float Wlds[DIN * 128];
    const int tid = threadIdx.x;

    // Build concatenated weight in LDS (tiny: DIN*128 elems, 2 MACs each)
    for (int idx = tid; idx < DIN * 128; idx += 256) {
        int k = idx >> 7, c = idx & 127;
        int r = c >> 5, o = c & 31;
        float v;
        if (r < NREL)
            v = comp[r * 2 + 0] * basis[(0 * DIN + k) * 32 + o]
              + comp[r * 2 + 1] * basis[(1 * DIN + k) * 32 + o];
        else
            v = root[k * 32 + o];
        Wlds[idx] = v;
    }
    __syncthreads();

    const int wave = tid >> 5;
    const int lane = tid & 31;
    const int tile = blockIdx.x * 8 + wave;          // 16-row tile index
    if (tile >= N_NODES / 16) return;                // wave-uniform exit
    const int row0 = tile * 16;
    const int half = lane >> 4;                      // K-half select
    const int ln   = lane & 15;
    const float* hrow = h + (size_t)(row0 + ln) * hstride;

    // Load all A fragments for this 16-row tile once (NK * v2f registers).
    constexpr int NK = DIN / 4;
    v2f afrag[NK];
#pragma unroll
    for (int kk = 0; kk < NK; ++kk) {
        const int k = kk * 4 + 2 * half;
        afrag[kk].x = hrow[k];
        afrag[kk].y = hrow[k + 1];
    }

#pragma unroll
    for (int ct = 0; ct < 8; ++ct) {                 // 8 column tiles of 16
        const int c0 = ct * 16;
        v8f acc = {};
#pragma unroll
        for (int kk = 0; kk < NK; ++kk) {
            const int k = kk * 4 + 2 * half;
            v2f b;
            b.x = Wlds[k * 128 + c0 + ln];
            b.y = Wlds[(k + 1) * 128 + c0 + ln];
            acc = __builtin_amdgcn_wmma_f32_16x16x4_f32(
                false, afrag[kk], false, b, (short)0, acc, false, false);
        }
        const int col   = c0 + ln;
        const int plane = col >> 5;
        const int o     = col & 31;
        float* outp = hw + ((size_t)plane * N_NODES + row0 + 8 * half) * 32 + o;
#pragma unroll
        for (int v = 0; v < 8; ++v)
            outp[(size_t)v * 32] = acc[v];           // row = row0 + 8*half + v
    }
}

// ---------------------------------------------------------------------------
// Edge-degree counting per (relation, dst); then invert: 1/max(cnt,1).
// Counts depend only on (edge_type, dst) -> computed once, reused by 4 layers.
// ---------------------------------------------------------------------------
__global__ void edge_count(const int* __restrict__ etype,
                           const int* __restrict__ dst,
                           float* __restrict__ cnt)
{
    int e = blockIdx.x * blockDim.x + threadIdx.x;
    if (e < NEDGE)
        atomicAdd(&cnt[(size_t)etype[e] * N_NODES + dst[e]], 1.0f);
}

__global__ void invert_count(float* __restrict__ cnt)
{
    int i = blockIdx.x * blockDim.x + threadIdx.x;
    if (i < NREL * N_NODES)
        cnt[i] = 1.0f / fmaxf(cnt[i], 1.0f);
}

// ---------------------------------------------------------------------------
// Edge scatter: one wave32 per edge, lane = feature.  Pre-scaling each message
// by 1/cnt(rel,dst) lets a single [N][32] accumulator compute
// sum_r (segment_sum_r / cnt_r) in one pass.
// ---------------------------------------------------------------------------
__global__ __launch_bounds__(256) void edge_scatter(
    const int* __restrict__ src, const int* __restrict__ dst,
    const int* __restrict__ etype,
    const float* __restrict__ hw, const float* __restrict__ invcnt,
    float* __restrict__ acc)
{
    const int wave = threadIdx.x >> 5;
    const int lane = threadIdx.x & 31;
    const int e = blockIdx.x * 8 + wave;
    if (e >= NEDGE) return;
    const int et = etype[e];
    const int s  = src[e];
    const int d  = dst[e];
    const float v = hw[((size_t)et * N_NODES + s) * 32 + lane];
    const float w = invcnt[(size_t)et * N_NODES + d];
    atomicAdd(&acc[(size_t)d * 32 + lane], v * w);
}

// ---------------------------------------------------------------------------
// combine: tanh(agg + h@root + bias) -> write into the concat states buffer
// ---------------------------------------------------------------------------
__global__ void rgcn_combine(const float* __restrict__ acc,
                             const float* __restrict__ hwroot,
                             const float* __restrict__ bias,
                             float* __restrict__ states_col)
{
    int i = blockIdx.x * blockDim.x + threadIdx.x;
    if (i >= N_NODES * 32) return;
    int n = i >> 5, o = i & 31;
    states_col[(size_t)n * 128 + o] = tanhf(acc[i] + hwroot[i] + bias[o]);
}

// ---------------------------------------------------------------------------
// Head MLP: f[g] = [states[user_g], states[bundle_g]] (user = g*100+1,
// bundle = g*100, structural).  16 graphs per block; 8 waves each own one
// 16-wide column tile of the 16x128 hidden layer via WMMA f32 over K=256.
// ---------------------------------------------------------------------------
__global__ __launch_bounds__(256) void mlp_head(
    const float* __restrict__ states, const float* __restrict__ w1,
    const float* __restrict__ b1, const float* __restrict__ w2,
    const float* __restrict__ b2, float* __restrict__ out)
{
    __shared__ float Alds[16 * 256];   // 16 KB: staged f rows
    __shared__ float Zlds[16 * 128];   // 8 KB: relu hidden
    const int tid = threadIdx.x;
    const int g0 = blockIdx.x * 16;

    for (int idx = tid; idx < 16 * 256; idx += 256) {
        int m = idx >> 8, k = idx & 255;
        int g = g0 + m;
        int node = (k < 128) ? (g * NPG + 1) : (g * NPG);  // user | bundle
        Alds[idx] = states[(size_t)node * 128 + (k & 127)];
    }
    __syncthreads();

    const int wave = tid >> 5, lane = tid & 31;
    const int half = lane >> 4, ln = lane & 15;
    const int c0 = wave * 16;
    v8f acc = {};
#pragma unroll 8
    for (int k0 = 0; k0 < 256; k0 += 4) {
        const int k = k0 + 2 * half;
        v2f a, b;
        a.x = Alds[ln * 256 + k];
        a.y = Alds[ln * 256 + k + 1];
        b.x = w1[(size_t)k * 128 + c0 + ln];
        b.y = w1[(size_t)(k + 1) * 128 + c0 + ln];
        acc = __builtin_amdgcn_wmma_f32_16x16x4_f32(
            false, a, false, b, (short)0, acc, false, false);
    }
    const int col = c0 + ln;
    const float bb = b1[col];
#pragma unroll
    for (int v = 0; v < 8; ++v) {
        int row = v + 8 * half;
        Zlds[row * 128 + col] = fmaxf(acc[v] + bb, 0.0f);
    }
    __syncthreads();
    if (tid < 16) {
        float s = 0.f;
        for (int c = 0; c < 128; ++c) s += Zlds[tid * 128 + c] * w2[c];
        s += b2[0];
        out[g0 + tid] = 1.0f / (1.0f + expf(-s));
    }
}

// ---------------------------------------------------------------------------
// d_in layout (pytree-leaf order of setup_inputs):
//  0 x [N,4] f32 | 1 edge_index [2,E] i32 | 2 edge_type [E] i32 | 3 num_graphs
//  4..7 basis[i] | 8..11 comp[i] | 12..15 root[i] | 16..19 bias[i]
//  20 w1 [256,128] | 21 b1 [128] | 22 w2 [128,1] | 23 b2 [1]
// ---------------------------------------------------------------------------
extern "C" void kernel_launch(void* const* d_in, const int* in_sizes, int n_in,
                              void* d_out, int out_size, void* d_ws, size_t ws_size,
                              hipStream_t stream)
{
    const float* x     = (const float*)d_in[0];
    const int*   eidx  = (const int*)d_in[1];
    const int*   etype = (const int*)d_in[2];
    const float* basis[4] = {(const float*)d_in[4],  (const float*)d_in[5],
                             (const float*)d_in[6],  (const float*)d_in[7]};
    const float* comp[4]  = {(const float*)d_in[8],  (const float*)d_in[9],
                             (const float*)d_in[10], (const float*)d_in[11]};
    const float* root[4]  = {(const float*)d_in[12], (const float*)d_in[13],
                             (const float*)d_in[14], (const float*)d_in[15]};
    const float* bias[4]  = {(const float*)d_in[16], (const float*)d_in[17],
                             (const float*)d_in[18], (const float*)d_in[19]};
    const float* w1 = (const float*)d_in[20];
    const float* b1 = (const float*)d_in[21];
    const float* w2 = (const float*)d_in[22];
    const float* b2 = (const float*)d_in[23];

    float* ws      = (float*)d_ws;
    float* hw      = ws;                                   // 4*N*32
    float* acc     = hw + (size_t)4 * N_NODES * 32;        // N*32
    float* invcnt  = acc + (size_t)N_NODES * 32;           // 3*N
    float* states  = invcnt + (size_t)NREL * N_NODES;      // N*128

    const int* src = eidx;
    const int* dst = eidx + NEDGE;

    // Per-(relation,dst) mean denominators, computed once.
    hipMemsetAsync(invcnt, 0, sizeof(float) * NREL * N_NODES, stream);
    edge_count<<<(NEDGE + 255) / 256, 256, 0, stream>>>(etype, dst, invcnt);
    invert_count<<<(NREL * N_NODES + 255) / 256, 256, 0, stream>>>(invcnt);

    const int tiles = N_NODES / 16;                        // 12500
    const int tblocks = (tiles + 7) / 8;                   // 8 waves/block

    for (int L = 0; L < 4; ++L) {
        if (L == 0)
            rgcn_transform<4><<<tblocks, 256, 0, stream>>>(
                x, 4, basis[0], comp[0], root[0], hw);
        else
            rgcn_transform<32><<<tblocks, 256, 0, stream>>>(
                states + (size_t)(L - 1) * 32, 128,
                basis[L], comp[L], root[L], hw);

        hipMemsetAsync(acc, 0, sizeof(float) * N_NODES * 32, stream);
        edge_scatter<<<NEDGE / 8, 256, 0, stream>>>(src, dst, etype, hw, invcnt, acc);
        rgcn_combine<<<((N_NODES * 32) + 255) / 256, 256, 0, stream>>>(
            acc, hw + (size_t)3 * N_NODES * 32, bias[L], states + (size_t)L * 32);
    }

    mlp_head<<<NGRAPH / 16, 256, 0, stream>>>(states, w1, b1, w2, b2, (float*)d_out);
}